// MakeGrid_36790689858049
// MI455X (gfx1250) — compile-verified
//
#include <hip/hip_runtime.h>
#include <stdint.h>

// Voxel scatter-add for MI455X (gfx1250, wave32).
//
// Reference semantics:
//   gc  = round_half_even((coords + 35.0) / 1.0)            [B,N,3] int
//   in  = all(0 <= gc < 71)
//   out = zeros(B,71,71,71,F); out[0, gcx, gcy, gcz, :] += in ? feat : 0
// (Faithful to original: ALL batches scatter into batch slice 0.)
//
// HBM-bound workload (366 MB zero-fill + 17.5 MB reads + 4.2M f32 atomics
// into a 45.8 MB region that is L2-resident). No matmul structure -> no WMMA.
// CDNA5 paths used: async global->LDS DMA (ASYNCcnt), wave32 shuffles,
// global_atomic_add_f32.

#define MAX_DIST 35.0f
#define BOX 71

// ---------------------------------------------------------------------------
// Kernel 1: zero the whole output grid with 128-bit stores.
// ---------------------------------------------------------------------------
__global__ void zero_f4(float* __restrict__ out, long long n) {
    long long i      = (long long)blockIdx.x * blockDim.x + threadIdx.x;
    long long stride = (long long)gridDim.x * blockDim.x;
    long long n4     = n >> 2;
    float4* out4 = reinterpret_cast<float4*>(out);
    for (long long k = i; k < n4; k += stride)
        out4[k] = make_float4(0.f, 0.f, 0.f, 0.f);
    for (long long k = (n4 << 2) + i; k < n; k += stride)   // tail (unused: n%4==0)
        out[k] = 0.f;
}

// ---------------------------------------------------------------------------
// Kernel 2 (fast path, F == 32): one wave32 per point, lane = feature index.
// Features are staged global->LDS with the CDNA5 async DMA path while the
// wave computes the voxel index, then scattered with coalesced f32 atomics.
// ---------------------------------------------------------------------------
__global__ __launch_bounds__(256) void scatter_f32x32(
    const float* __restrict__ coords,    // [P,3]
    const float* __restrict__ features,  // [P,32]
    float* __restrict__ out,             // [B,71,71,71,32], batch 0 only
    int P) {
    __shared__ float buf[256];

    const int tid  = threadIdx.x;
    const int lane = tid & 31;
    const int wave = tid >> 5;
    const int p    = blockIdx.x * 8 + wave;   // 8 waves/block = 8 points/block
    if (p >= P) return;                        // wave-uniform exit

    // --- async DMA: features[p*32 + lane] -> LDS buf[tid] (4 bytes/lane) ---
    // GVS mode: mem_addr = SADDR(64b sgpr pair) + VADDR(32b vgpr byte offset).
    // LDS destination address is taken from the VDST vgpr; low 32 bits of a flat shared pointer are
    // exactly the LDS offset the async unit expects.
    uint32_t ldsAddr = (uint32_t)(uintptr_t)(&buf[tid]);
    uint32_t gOff    = (uint32_t)(((unsigned)p * 32u + (unsigned)lane) * 4u);
    asm volatile("global_load_async_to_lds_b32 %0, %1, %2"
                 :
                 : "v"(ldsAddr), "v"(gOff), "s"(features)
                 : "memory");

    // --- overlap: voxel index computation (independent of the DMA) ---
    float c = (lane < 3) ? coords[p * 3 + lane] : 0.0f;
    float x = __shfl(c, 0, 32);
    float y = __shfl(c, 1, 32);
    float z = __shfl(c, 2, 32);

    // GRID_RES == 1.0 -> divide is exact no-op; round-to-nearest-even matches
    // jnp.round.
    int ix = __float2int_rn(x + MAX_DIST);
    int iy = __float2int_rn(y + MAX_DIST);
    int iz = __float2int_rn(z + MAX_DIST);
    bool inbox = ((unsigned)ix < BOX) & ((unsigned)iy < BOX) & ((unsigned)iz < BOX);

    // --- drain this wave's async DMA, then scatter from LDS ---
    asm volatile("s_wait_asynccnt 0x0" ::: "memory");
    float f = buf[tid];

    if (inbox) {
        int idx = ((ix * BOX + iy) * BOX + iz) * 32 + lane;  // batch slice 0
        atomicAdd(out + idx, f);   // global_atomic_add_f32, 32 coalesced lanes
    }
}

// ---------------------------------------------------------------------------
// Generic fallback (any F): one thread per (point, feature).
// ---------------------------------------------------------------------------
__global__ void scatter_generic(
    const float* __restrict__ coords,
    const float* __restrict__ features,
    float* __restrict__ out,
    long long P, int F) {
    long long gid   = (long long)blockIdx.x * blockDim.x + threadIdx.x;
    long long total = P * (long long)F;
    if (gid >= total) return;
    long long p = gid / F;
    int       f = (int)(gid - p * F);

    float x = coords[p * 3 + 0];
    float y = coords[p * 3 + 1];
    float z = coords[p * 3 + 2];
    int ix = __float2int_rn(x + MAX_DIST);
    int iy = __float2int_rn(y + MAX_DIST);
    int iz = __float2int_rn(z + MAX_DIST);
    if (((unsigned)ix < BOX) & ((unsigned)iy < BOX) & ((unsigned)iz < BOX)) {
        long long idx = (((long long)ix * BOX + iy) * BOX + iz) * F + f;
        atomicAdd(out + idx, features[p * F + f]);
    }
}

// ---------------------------------------------------------------------------
extern "C" void kernel_launch(void* const* d_in, const int* in_sizes, int n_in,
                              void* d_out, int out_size, void* d_ws, size_t ws_size,
                              hipStream_t stream) {
    (void)n_in; (void)d_ws; (void)ws_size;

    const float* coords   = (const float*)d_in[0];   // [B,N,3] f32
    const float* features = (const float*)d_in[1];   // [B,N,F] f32
    float*       out      = (float*)d_out;           // [B,71,71,71,F] f32

    const long long P = (long long)in_sizes[0] / 3;              // B*N points
    const int       F = (P > 0) ? (int)(in_sizes[1] / P) : 0;

    // 1) zero-fill the full output (harness poisons it with 0xAA).
    zero_f4<<<4096, 256, 0, stream>>>(out, (long long)out_size);

    // 2) scatter-add into batch slice 0.
    if (F == 32) {
        int blocks = (int)((P + 7) / 8);                 // 8 points per block
        scatter_f32x32<<<blocks, 256, 0, stream>>>(coords, features, out, (int)P);
    } else if (F > 0) {
        long long total  = P * (long long)F;
        int       blocks = (int)((total + 255) / 256);
        scatter_generic<<<blocks, 256, 0, stream>>>(coords, features, out, P, F);
    }
}